// AttentionOneHotConv_85358180041521
// MI455X (gfx1250) — compile-verified
//
#include <hip/hip_runtime.h>
#include <hip/hip_bf16.h>
#include <math.h>

// Problem constants (from reference)
#define NNODE 2048
#define NE    16384
#define EP    (NE + NNODE)   // 18432 edges incl. self loops
#define FIN   128
#define FOUT  128
#define NH    4
#define OHF   8
#define KCAT  (FIN + OHF)    // 136
#define HXO   (NH * FOUT)    // 512
#define NEG_SLOPE 0.2f

typedef float v2f __attribute__((ext_vector_type(2)));
typedef float v8f __attribute__((ext_vector_type(8)));

__device__ __forceinline__ float symlogf(float v) {
    return copysignf(log1pf(fabsf(v)), v);
}

__device__ __forceinline__ void atomicMaxFloat(float* addr, float val) {
    unsigned int* uaddr = (unsigned int*)addr;
    unsigned int cur = *uaddr;
    while (__uint_as_float(cur) < val) {
        unsigned int prev = atomicCAS(uaddr, cur, __float_as_uint(val));
        if (prev == cur) break;
        cur = prev;
    }
}

// ---------------- K1: L = symlog(onehot) (elementwise) ----------------
__global__ void symlog_kernel(const float* __restrict__ onehot, float* __restrict__ L) {
    size_t i = (size_t)blockIdx.x * blockDim.x + threadIdx.x;
    size_t total = (size_t)NNODE * NNODE;
    if (i < total) L[i] = symlogf(onehot[i]);
}

// ---------------- K2: per-row bitonic sort (ascending) ----------------
__global__ void __launch_bounds__(256) sort_row_kernel(const float* __restrict__ L,
                                                       float* __restrict__ prep) {
    __shared__ float s[NNODE];
    int tid = threadIdx.x;
    const float* rowp = L + (size_t)blockIdx.x * NNODE;
    for (int i = tid; i < NNODE; i += 256) s[i] = rowp[i];
    __syncthreads();
    for (int k = 2; k <= NNODE; k <<= 1) {
        for (int j = k >> 1; j > 0; j >>= 1) {
            for (int i = tid; i < NNODE; i += 256) {
                int ixj = i ^ j;
                if (ixj > i) {
                    float a = s[i], b = s[ixj];
                    bool up = ((i & k) == 0);
                    if ((a > b) == up) { s[i] = b; s[ixj] = a; }
                }
            }
            __syncthreads();
        }
    }
    float* op = prep + (size_t)blockIdx.x * NNODE;
    for (int i = tid; i < NNODE; i += 256) op[i] = s[i];
}

// ---------------- K3: fused conv1->relu->conv2->relu->meanpool->fc ----------------
__global__ void __launch_bounds__(256) conv_pipeline_kernel(
        const float* __restrict__ prep,
        const float* __restrict__ c1w, const float* __restrict__ c1b,
        const float* __restrict__ c2w, const float* __restrict__ c2b,
        const float* __restrict__ fcw, const float* __restrict__ fcb,
        float* __restrict__ oh_feat) {
    __shared__ float row[NNODE];
    __shared__ float w1[24], b1[8], w2[384], b2[16], wf[128], bf[8];
    __shared__ float red[8][16];
    __shared__ float pooled[16];
    int tid = threadIdx.x;
    int r = blockIdx.x;
    if (tid < 24)  w1[tid] = c1w[tid];
    if (tid < 8)   b1[tid] = c1b[tid];
    for (int i = tid; i < 384; i += 256) w2[i] = c2w[i];
    if (tid < 16)  b2[tid] = c2b[tid];
    if (tid < 128) wf[tid] = fcw[tid];
    if (tid < 8)   bf[tid] = fcb[tid];
    for (int i = tid; i < NNODE; i += 256) row[i] = prep[(size_t)r * NNODE + i];
    __syncthreads();

    float acc[16];
#pragma unroll
    for (int c = 0; c < 16; ++c) acc[c] = 0.f;

    for (int p = tid; p < NNODE; p += 256) {
        float in[5];
#pragma unroll
        for (int t = 0; t < 5; ++t) {
            int q = p - 2 + t;
            in[t] = (q >= 0 && q < NNODE) ? row[q] : 0.f;
        }
        float h1[3][8];
#pragma unroll
        for (int dq = 0; dq < 3; ++dq) {
            int q = p - 1 + dq;
            bool valid = (q >= 0 && q < NNODE);
#pragma unroll
            for (int c = 0; c < 8; ++c) {
                float v = b1[c];
#pragma unroll
                for (int t = 0; t < 3; ++t) v = fmaf(w1[c * 3 + t], in[dq + t], v);
                h1[dq][c] = valid ? fmaxf(v, 0.f) : 0.f;
            }
        }
#pragma unroll
        for (int oc = 0; oc < 16; ++oc) {
            float v = b2[oc];
#pragma unroll
            for (int ic = 0; ic < 8; ++ic)
#pragma unroll
                for (int t = 0; t < 3; ++t)
                    v = fmaf(w2[(oc * 8 + ic) * 3 + t], h1[t][ic], v);
            acc[oc] += fmaxf(v, 0.f);
        }
    }
    // wave32 reduction then cross-wave via LDS
#pragma unroll
    for (int c = 0; c < 16; ++c)
        for (int off = 16; off > 0; off >>= 1)
            acc[c] += __shfl_xor(acc[c], off, 32);
    int wave = tid >> 5, lane = tid & 31;
    if (lane == 0)
        for (int c = 0; c < 16; ++c) red[wave][c] = acc[c];
    __syncthreads();
    if (tid < 16) {
        float s = 0.f;
        for (int w = 0; w < 8; ++w) s += red[w][tid];
        pooled[tid] = s * (1.0f / (float)NNODE);
    }
    __syncthreads();
    if (tid < OHF) {
        float v = bf[tid];
        for (int c = 0; c < 16; ++c) v = fmaf(wf[tid * 16 + c], pooled[c], v);
        oh_feat[(size_t)r * OHF + tid] = v;
    }
}

// ---------------- K4: WMMA fp32 GEMM  xh = [x | oh_feat] @ lin_w^T ----------------
// M=2048, K=136, N=512.  One 16x16 tile per wave32, K stepped by 4 via
// V_WMMA_F32_16X16X4_F32.  A(16x4): lanes0-15 M, VGPR{0,1}=K{0,1}; lanes16-31 K{2,3}.
__global__ void __launch_bounds__(256) gemm_wmma_kernel(
        const float* __restrict__ x, const float* __restrict__ ohf,
        const float* __restrict__ w, float* __restrict__ xh) {
    int wave = (blockIdx.x * blockDim.x + threadIdx.x) >> 5;
    int lane = threadIdx.x & 31;
    int tm = wave >> 5;          // 128 M-tiles
    int tn = wave & 31;          // 32  N-tiles
    int m0 = tm * 16, n0 = tn * 16;
    int half = lane >> 4;        // selects K pair {0,1} vs {2,3}
    int lr = lane & 15;
    v8f c = {};
    int m = m0 + lr;
    int n = n0 + lr;
    for (int k0 = 0; k0 < KCAT; k0 += 4) {
        int ka = k0 + half * 2;
        v2f a, b;
        a.x = (ka     < FIN) ? x[m * FIN + ka]     : ohf[m * OHF + (ka - FIN)];
        a.y = (ka + 1 < FIN) ? x[m * FIN + ka + 1] : ohf[m * OHF + (ka + 1 - FIN)];
        b.x = w[n * KCAT + ka];
        b.y = w[n * KCAT + ka + 1];
        c = __builtin_amdgcn_wmma_f32_16x16x4_f32(false, a, false, b,
                                                  (short)0, c, false, false);
    }
#pragma unroll
    for (int r = 0; r < 8; ++r) {
        int mm = m0 + r + half * 8;
        xh[(size_t)mm * HXO + n0 + lr] = c[r];
    }
}

// ---------------- K5: per-head attention logits sa/ra ----------------
__global__ void __launch_bounds__(256) attn_logits_kernel(
        const float* __restrict__ xh, const float* __restrict__ att_l,
        const float* __restrict__ att_r, float* __restrict__ sa, float* __restrict__ ra) {
    int gw = (blockIdx.x * blockDim.x + threadIdx.x) >> 5;
    int lane = threadIdx.x & 31;
    int i = gw >> 2, h = gw & 3;
    const float* base = xh + (size_t)i * HXO + h * FOUT;
    float sl = 0.f, sr = 0.f;
    for (int o = lane; o < FOUT; o += 32) {
        float v = base[o];
        sl = fmaf(v, att_l[h * FOUT + o], sl);
        sr = fmaf(v, att_r[h * FOUT + o], sr);
    }
    for (int off = 16; off > 0; off >>= 1) {
        sl += __shfl_xor(sl, off, 32);
        sr += __shfl_xor(sr, off, 32);
    }
    if (lane == 0) { sa[i * NH + h] = sl; ra[i * NH + h] = sr; }
}

// ---------------- K6: init segment-softmax stats ----------------
__global__ void init_stats_kernel(float* __restrict__ amax, float* __restrict__ den) {
    int i = blockIdx.x * blockDim.x + threadIdx.x;
    if (i < NNODE * NH) { amax[i] = -3.402823466e+38f; den[i] = 0.f; }
}

// ---------------- K7: per-edge leaky-relu logits * dotatt ----------------
__global__ void __launch_bounds__(256) edge_alpha_kernel(
        const float* __restrict__ L, const float* __restrict__ sa,
        const float* __restrict__ ra, const int* __restrict__ src,
        const int* __restrict__ dst, float* __restrict__ alpha) {
    int gw = (blockIdx.x * blockDim.x + threadIdx.x) >> 5;
    int lane = threadIdx.x & 31;
    if (gw >= EP) return;
    int s = (gw < NE) ? src[gw] : gw - NE;
    int d = (gw < NE) ? dst[gw] : gw - NE;
    const float4* Ls = (const float4*)(L + (size_t)s * NNODE);
    const float4* Ld = (const float4*)(L + (size_t)d * NNODE);
    float dot = 0.f;
#pragma unroll 4
    for (int it = 0; it < NNODE / 128; ++it) {      // 16 iters of float4 per lane
        float4 av = Ls[it * 32 + lane];
        float4 bv = Ld[it * 32 + lane];
        dot = fmaf(av.x, bv.x, dot); dot = fmaf(av.y, bv.y, dot);
        dot = fmaf(av.z, bv.z, dot); dot = fmaf(av.w, bv.w, dot);
    }
    for (int off = 16; off > 0; off >>= 1) dot += __shfl_xor(dot, off, 32);
    float da = 1.f / (dot + 1.f);
    if (lane < NH) {
        float v = sa[s * NH + lane] + ra[d * NH + lane];
        v = (v > 0.f) ? v : NEG_SLOPE * v;
        alpha[(size_t)gw * NH + lane] = v * da;
    }
}

// ---------------- K8: segment max over destination ----------------
__global__ void seg_max_kernel(const float* __restrict__ alpha,
                               const int* __restrict__ dst, float* __restrict__ amax) {
    int idx = blockIdx.x * blockDim.x + threadIdx.x;
    if (idx >= EP * NH) return;
    int e = idx >> 2, h = idx & 3;
    int d = (e < NE) ? dst[e] : e - NE;
    atomicMaxFloat(&amax[d * NH + h], alpha[idx]);
}

// ---------------- K9: exp(alpha - amax), segment-sum denominators ----------------
__global__ void seg_exp_kernel(float* __restrict__ alpha, const int* __restrict__ dst,
                               const float* __restrict__ amax, float* __restrict__ den) {
    int idx = blockIdx.x * blockDim.x + threadIdx.x;
    if (idx >= EP * NH) return;
    int e = idx >> 2, h = idx & 3;
    int d = (e < NE) ? dst[e] : e - NE;
    float ex = __expf(alpha[idx] - amax[d * NH + h]);
    alpha[idx] = ex;
    atomicAdd(&den[d * NH + h], ex);
}

// ---------------- K10: init outputs (new_x = bias, new_onehot = onehot) ----------------
__global__ void init_out_kernel(const float* __restrict__ onehot,
                                const float* __restrict__ bias,
                                float* __restrict__ out_x, float* __restrict__ out_oh) {
    size_t stride = (size_t)gridDim.x * blockDim.x;
    size_t i0 = (size_t)blockIdx.x * blockDim.x + threadIdx.x;
    for (size_t t = i0; t < (size_t)NNODE * NNODE; t += stride) out_oh[t] = onehot[t];
    for (size_t t = i0; t < (size_t)NNODE * HXO; t += stride) out_x[t] = bias[t % HXO];
}

// ---------------- K11: normalized scatter-aggregate ----------------
__global__ void __launch_bounds__(256) aggregate_kernel(
        const float* __restrict__ alpha, const float* __restrict__ den,
        const float* __restrict__ xh, const float* __restrict__ onehot,
        const int* __restrict__ src, const int* __restrict__ dst,
        float* __restrict__ out_x, float* __restrict__ out_oh) {
    __shared__ float wsh[NH];
    int e = blockIdx.x;
    int tid = threadIdx.x;
    int s = (e < NE) ? src[e] : e - NE;
    int d = (e < NE) ? dst[e] : e - NE;
    if (tid < NH)
        wsh[tid] = alpha[(size_t)e * NH + tid] / (den[d * NH + tid] + 1e-16f);
    __syncthreads();
    const float* xrow = xh + (size_t)s * HXO;
    float* orow = out_x + (size_t)d * HXO;
    for (int col = tid; col < HXO; col += 256) {
        int h = col >> 7;
        atomicAdd(&orow[col], xrow[col] * wsh[h]);
    }
    const float* srow = onehot + (size_t)s * NNODE;
    float* drow = out_oh + (size_t)d * NNODE;
    for (int col = tid; col < NNODE; col += 256)
        atomicAdd(&drow[col], srow[col]);
}

extern "C" void kernel_launch(void* const* d_in, const int* in_sizes, int n_in,
                              void* d_out, int out_size, void* d_ws, size_t ws_size,
                              hipStream_t stream) {
    const float* x      = (const float*)d_in[0];
    const float* onehot = (const float*)d_in[1];
    const int*   src    = (const int*)d_in[2];
    const int*   dst    = (const int*)d_in[3];
    const float* lin_w  = (const float*)d_in[4];
    const float* att_l  = (const float*)d_in[5];
    const float* att_r  = (const float*)d_in[6];
    const float* bias   = (const float*)d_in[7];
    const float* c1w    = (const float*)d_in[8];
    const float* c1b    = (const float*)d_in[9];
    const float* c2w    = (const float*)d_in[10];
    const float* c2b    = (const float*)d_in[11];
    const float* fcw    = (const float*)d_in[12];
    const float* fcb    = (const float*)d_in[13];

    float* out_x  = (float*)d_out;
    float* out_oh = out_x + (size_t)NNODE * HXO;

    // Workspace carve-up (~36.5 MB total)
    char* ws = (char*)d_ws;
    float* L    = (float*)ws;                ws += (size_t)NNODE * NNODE * sizeof(float);
    float* prep = (float*)ws;                ws += (size_t)NNODE * NNODE * sizeof(float);
    float* xh   = (float*)ws;                ws += (size_t)NNODE * HXO * sizeof(float);
    float* ohft = (float*)ws;                ws += (size_t)NNODE * OHF * sizeof(float);
    float* sa   = (float*)ws;                ws += (size_t)NNODE * NH * sizeof(float);
    float* ra   = (float*)ws;                ws += (size_t)NNODE * NH * sizeof(float);
    float* amax = (float*)ws;                ws += (size_t)NNODE * NH * sizeof(float);
    float* den  = (float*)ws;                ws += (size_t)NNODE * NH * sizeof(float);
    float* alpha = (float*)ws;               ws += (size_t)EP * NH * sizeof(float);

    // 1) L = symlog(onehot)
    {
        size_t total = (size_t)NNODE * NNODE;
        symlog_kernel<<<(unsigned)((total + 255) / 256), 256, 0, stream>>>(onehot, L);
    }
    // 2) prep = sort of symlog rows (symlog monotonic => sort(L) == symlog(sort(onehot)))
    sort_row_kernel<<<NNODE, 256, 0, stream>>>(L, prep);
    // 3) oh_feat
    conv_pipeline_kernel<<<NNODE, 256, 0, stream>>>(prep, c1w, c1b, c2w, c2b, fcw, fcb, ohft);
    // 4) xh = [x | oh_feat] @ lin_w^T   (WMMA f32 16x16x4)
    gemm_wmma_kernel<<<(NNODE / 16) * (HXO / 16) / 8, 256, 0, stream>>>(x, ohft, lin_w, xh);
    // 5) sa / ra
    attn_logits_kernel<<<NNODE * NH / 8, 256, 0, stream>>>(xh, att_l, att_r, sa, ra);
    // 6) init softmax stats
    init_stats_kernel<<<(NNODE * NH + 255) / 256, 256, 0, stream>>>(amax, den);
    // 7) alpha = leaky_relu(sa[s]+ra[d]) * dotatt
    edge_alpha_kernel<<<EP / 8, 256, 0, stream>>>(L, sa, ra, src, dst, alpha);
    // 8) segment max
    seg_max_kernel<<<(EP * NH + 255) / 256, 256, 0, stream>>>(alpha, dst, amax);
    // 9) exp + denominator
    seg_exp_kernel<<<(EP * NH + 255) / 256, 256, 0, stream>>>(alpha, dst, amax, den);
    // 10) init outputs
    init_out_kernel<<<4096, 256, 0, stream>>>(onehot, bias, out_x, out_oh);
    // 11) scatter aggregate
    aggregate_kernel<<<EP, 256, 0, stream>>>(alpha, den, xh, onehot, src, dst, out_x, out_oh);
}